// SpikeNet_7602092114108
// MI455X (gfx1250) — compile-verified
//
#include <hip/hip_runtime.h>

// ---------------- problem constants (from reference) ----------------
#define F      128
#define N0     4096
#define S1     5
#define S2     2
#define H1     128
#define H2     128
#define TSTEPS 5
#define OUTC   64
#define N1     (N0 * S1)        // 20480
#define NSELF  (N0 + N1)        // 24576
#define KHEAD  (TSTEPS * H2)    // 640

typedef __attribute__((ext_vector_type(2))) float v2f;
typedef __attribute__((ext_vector_type(8))) float v8f;

// D = A(16x4,f32) * B(4x16,f32) + C(16x16,f32)  -> v_wmma_f32_16x16x4_f32
__device__ __forceinline__ v8f wmma4(v2f a, v2f b, v8f c) {
    return __builtin_amdgcn_wmma_f32_16x16x4_f32(
        false, a, false, b, (short)0, c, false, false);
}

// LDS row stride in dwords: 128 + 4 pad -> bank = (4*m + k) mod 64; the 16
// row-lanes of each ds_load_b64 half land on distinct banks.
#define LSTR 132

// CDNA5 async copy: 16B global -> LDS, no VGPR round trip, tracked by ASYNCcnt.
__device__ __forceinline__ void async_copy16(const float* g, float* lds_dst) {
    unsigned ldsoff = (unsigned)(unsigned long long)lds_dst;   // low 32b of generic = LDS offset
    unsigned long long ga = (unsigned long long)g;
    asm volatile("global_load_async_to_lds_b128 %0, %1, off"
                 :: "v"(ldsoff), "v"(ga) : "memory");
}
__device__ __forceinline__ void wait_async0() {
    asm volatile("s_wait_asynccnt 0" ::: "memory");
}

// ---------------------------------------------------------------------------
// Weight transpose: WT[n*K + k] = W[k*N + n]  (done once per launch; weights
// are tiny and L2-resident, this makes every B fragment a contiguous b64).
// ---------------------------------------------------------------------------
__global__ __launch_bounds__(256) void transpose_kernel(
    const float* __restrict__ W, float* __restrict__ WT, int K, int N)
{
    const int i = blockIdx.x * blockDim.x + threadIdx.x;
    if (i < K * N) {
        const int k = i / N;
        const int n = i - k * N;
        WT[(size_t)n * K + k] = W[i];
    }
}

// ---------------------------------------------------------------------------
// Layer 0: gather rows of x (self rows via async->LDS), mean-pool neighbors,
// dual GEMM (F->H1) per 32-row block; 8 waves, each owns a 16x32 output slab.
// ---------------------------------------------------------------------------
__global__ __launch_bounds__(256) void layer0_kernel(
    const float* __restrict__ x,
    const int*   __restrict__ nodes,
    const int*   __restrict__ nbr1_t,   // N1 indices for this t
    const int*   __restrict__ nbr2_t,   // N1*S2 indices for this t
    const float* __restrict__ WrT0,     // H1 x F (transposed)
    const float* __restrict__ WlT0,     // H1 x F (transposed)
    const float* __restrict__ br0,
    const float* __restrict__ bl0,
    float*       __restrict__ s0)       // NSELF x H1
{
    __shared__ float As[32 * LSTR];
    __shared__ float An[32 * LSTR];

    const int tid = threadIdx.x;
    const int R   = blockIdx.x * 32;

    // ---- gather + pool: thread -> (row = tid/8, 16 floats) ----
    {
        const int row = tid >> 3;          // 0..31
        const int c16 = (tid & 7) * 16;    // col base
        const int gr  = R + row;

        int selfIdx, nsum;
        int nIdx[S1] = {0, 0, 0, 0, 0};
        float scale;
        if (R < N0) {                       // uniform per block (N0 % 32 == 0)
            selfIdx = nodes[gr];
            nsum = S1; scale = 1.0f / (float)S1;
            #pragma unroll
            for (int s = 0; s < S1; ++s) nIdx[s] = nbr1_t[gr * S1 + s];
        } else {
            const int q = gr - N0;
            selfIdx = nbr1_t[q];
            nsum = S2; scale = 1.0f / (float)S2;
            #pragma unroll
            for (int s = 0; s < S2; ++s) nIdx[s] = nbr2_t[q * S2 + s];
        }

        // self row: pure copy -> async direct to LDS (4 x b128)
        const float* srow = x + (size_t)selfIdx * F + c16;
        float*       sdst = &As[row * LSTR + c16];
        #pragma unroll
        for (int i = 0; i < 4; ++i) async_copy16(srow + 4 * i, sdst + 4 * i);

        // neighbor rows: need arithmetic, go through VGPRs
        float acc[16];
        #pragma unroll
        for (int i = 0; i < 16; ++i) acc[i] = 0.0f;
        for (int s = 0; s < nsum; ++s) {
            const float4* np = (const float4*)(x + (size_t)nIdx[s] * F + c16);
            #pragma unroll
            for (int i = 0; i < 4; ++i) {
                float4 v = np[i];
                acc[4*i+0] += v.x; acc[4*i+1] += v.y;
                acc[4*i+2] += v.z; acc[4*i+3] += v.w;
            }
        }
        float* nd = &An[row * LSTR + c16];
        #pragma unroll
        for (int i = 0; i < 4; ++i) {
            *(float4*)(nd + 4 * i) = make_float4(acc[4*i+0]*scale, acc[4*i+1]*scale,
                                                 acc[4*i+2]*scale, acc[4*i+3]*scale);
        }
    }
    wait_async0();
    __syncthreads();

    // ---- WMMA: wave -> (row tile rt, 32-col slab) ----
    const int wave = tid >> 5;
    const int lane = tid & 31;
    const int rt   = wave >> 2;         // 0/1
    const int n0   = (wave & 3) * 32;   // col slab base
    const int m    = lane & 15;
    const int kh   = lane >> 4;         // lanes>=16 hold K+2,K+3
    const int ar   = rt * 16 + m;

    v8f acc0 = {0.f,0.f,0.f,0.f,0.f,0.f,0.f,0.f};
    v8f acc1 = {0.f,0.f,0.f,0.f,0.f,0.f,0.f,0.f};

    #pragma unroll 4
    for (int k = 0; k < F; k += 4) {
        const int k0 = k + 2 * kh;
        v2f aS  = *(const v2f*)&As[ar * LSTR + k0];
        v2f aN  = *(const v2f*)&An[ar * LSTR + k0];
        v2f bR0 = *(const v2f*)(WrT0 + (size_t)(n0 + m)      * F + k0);
        v2f bR1 = *(const v2f*)(WrT0 + (size_t)(n0 + 16 + m) * F + k0);
        v2f bL0 = *(const v2f*)(WlT0 + (size_t)(n0 + m)      * F + k0);
        v2f bL1 = *(const v2f*)(WlT0 + (size_t)(n0 + 16 + m) * F + k0);
        acc0 = wmma4(aS, bR0, acc0);
        acc0 = wmma4(aN, bL0, acc0);
        acc1 = wmma4(aS, bR1, acc1);
        acc1 = wmma4(aN, bL1, acc1);
    }

    const int col0 = n0 + m, col1 = n0 + 16 + m;
    const float b0 = br0[col0] + bl0[col0];
    const float b1 = br0[col1] + bl0[col1];
    #pragma unroll
    for (int j = 0; j < 8; ++j) {
        const int row = R + rt * 16 + j + 8 * kh;
        s0[(size_t)row * H1 + col0] = (acc0[j] + b0 >= 1.0f) ? 1.0f : 0.0f;
        s0[(size_t)row * H1 + col1] = (acc1[j] + b1 >= 1.0f) ? 1.0f : 0.0f;
    }
}

// ---------------------------------------------------------------------------
// Layer 1: pool s0, dual GEMM (H1->H2), spike -> spikes column block t.
// Same 32-row / 8-wave structure; contiguous self rows go async->LDS.
// ---------------------------------------------------------------------------
__global__ __launch_bounds__(256) void layer1_kernel(
    const float* __restrict__ s0,       // NSELF x H1
    const float* __restrict__ WrT1,     // transposed
    const float* __restrict__ WlT1,     // transposed
    const float* __restrict__ br1,
    const float* __restrict__ bl1,
    float*       __restrict__ spikes,   // N0 x KHEAD
    int t)
{
    __shared__ float As[32 * LSTR];
    __shared__ float An[32 * LSTR];

    const int tid = threadIdx.x;
    const int R   = blockIdx.x * 32;

    {
        const int row = tid >> 3;
        const int c16 = (tid & 7) * 16;
        const int gr  = R + row;

        const float* srow = s0 + (size_t)gr * H1 + c16;
        float*       sdst = &As[row * LSTR + c16];
        #pragma unroll
        for (int i = 0; i < 4; ++i) async_copy16(srow + 4 * i, sdst + 4 * i);

        float acc[16];
        #pragma unroll
        for (int i = 0; i < 16; ++i) acc[i] = 0.0f;
        #pragma unroll
        for (int s = 0; s < S1; ++s) {
            const float4* np =
                (const float4*)(s0 + (size_t)(N0 + gr * S1 + s) * H1 + c16);
            #pragma unroll
            for (int i = 0; i < 4; ++i) {
                float4 v = np[i];
                acc[4*i+0] += v.x; acc[4*i+1] += v.y;
                acc[4*i+2] += v.z; acc[4*i+3] += v.w;
            }
        }
        const float scale = 1.0f / (float)S1;
        float* nd = &An[row * LSTR + c16];
        #pragma unroll
        for (int i = 0; i < 4; ++i) {
            *(float4*)(nd + 4 * i) = make_float4(acc[4*i+0]*scale, acc[4*i+1]*scale,
                                                 acc[4*i+2]*scale, acc[4*i+3]*scale);
        }
    }
    wait_async0();
    __syncthreads();

    const int wave = tid >> 5;
    const int lane = tid & 31;
    const int rt   = wave >> 2;
    const int n0   = (wave & 3) * 32;
    const int m    = lane & 15;
    const int kh   = lane >> 4;
    const int ar   = rt * 16 + m;

    v8f acc0 = {0.f,0.f,0.f,0.f,0.f,0.f,0.f,0.f};
    v8f acc1 = {0.f,0.f,0.f,0.f,0.f,0.f,0.f,0.f};

    #pragma unroll 4
    for (int k = 0; k < H1; k += 4) {
        const int k0 = k + 2 * kh;
        v2f aS  = *(const v2f*)&As[ar * LSTR + k0];
        v2f aN  = *(const v2f*)&An[ar * LSTR + k0];
        v2f bR0 = *(const v2f*)(WrT1 + (size_t)(n0 + m)      * H1 + k0);
        v2f bR1 = *(const v2f*)(WrT1 + (size_t)(n0 + 16 + m) * H1 + k0);
        v2f bL0 = *(const v2f*)(WlT1 + (size_t)(n0 + m)      * H1 + k0);
        v2f bL1 = *(const v2f*)(WlT1 + (size_t)(n0 + 16 + m) * H1 + k0);
        acc0 = wmma4(aS, bR0, acc0);
        acc0 = wmma4(aN, bL0, acc0);
        acc1 = wmma4(aS, bR1, acc1);
        acc1 = wmma4(aN, bL1, acc1);
    }

    const int col0 = n0 + m, col1 = n0 + 16 + m;
    const float b0 = br1[col0] + bl1[col0];
    const float b1 = br1[col1] + bl1[col1];
    #pragma unroll
    for (int j = 0; j < 8; ++j) {
        const int row = R + rt * 16 + j + 8 * kh;
        float* dst = spikes + (size_t)row * KHEAD + t * H2;
        dst[col0] = (acc0[j] + b0 >= 1.0f) ? 1.0f : 0.0f;
        dst[col1] = (acc1[j] + b1 >= 1.0f) ? 1.0f : 0.0f;
    }
}

// ---------------------------------------------------------------------------
// Head: (N0 x 640) @ (640 x 64) + bp. 16 rows x 64 cols per block, 4 waves.
// A fragments straight from L2-resident spikes; B from transposed Wp (b64).
// ---------------------------------------------------------------------------
__global__ __launch_bounds__(128) void head_kernel(
    const float* __restrict__ spikes,   // N0 x KHEAD
    const float* __restrict__ WpT,      // OUTC x KHEAD (transposed)
    const float* __restrict__ bp,
    float*       __restrict__ out)      // N0 x OUTC
{
    const int tid  = threadIdx.x;
    const int R    = blockIdx.x * 16;
    const int wave = tid >> 5;          // 0..3
    const int lane = tid & 31;
    const int m    = lane & 15;
    const int kh   = lane >> 4;
    const int n0c  = wave * 16;

    v8f acc = {0.f,0.f,0.f,0.f,0.f,0.f,0.f,0.f};

    #pragma unroll 4
    for (int k = 0; k < KHEAD; k += 4) {
        const int k0 = k + 2 * kh;
        v2f a = *(const v2f*)(spikes + (size_t)(R + m) * KHEAD + k0);
        v2f b = *(const v2f*)(WpT + (size_t)(n0c + m) * KHEAD + k0);
        acc = wmma4(a, b, acc);
    }

    const int   col  = n0c + m;
    const float bias = bp[col];
    #pragma unroll
    for (int j = 0; j < 8; ++j) {
        const int row = j + 8 * kh;
        out[(size_t)(R + row) * OUTC + col] = acc[j] + bias;
    }
}

// ---------------------------------------------------------------------------
extern "C" void kernel_launch(void* const* d_in, const int* in_sizes, int n_in,
                              void* d_out, int out_size, void* d_ws, size_t ws_size,
                              hipStream_t stream)
{
    const float* x     = (const float*)d_in[0];
    const int*   nodes = (const int*)  d_in[1];
    const int*   nbr1  = (const int*)  d_in[2];   // T x N1
    const int*   nbr2  = (const int*)  d_in[3];   // T x N1*S2
    const float* Wl0   = (const float*)d_in[4];
    const float* bl0   = (const float*)d_in[5];
    const float* Wr0   = (const float*)d_in[6];
    const float* br0   = (const float*)d_in[7];
    const float* Wl1   = (const float*)d_in[8];
    const float* bl1   = (const float*)d_in[9];
    const float* Wr1   = (const float*)d_in[10];
    const float* br1   = (const float*)d_in[11];
    const float* Wp    = (const float*)d_in[12];
    const float* bp    = (const float*)d_in[13];
    float*       out   = (float*)d_out;

    // workspace layout (floats)
    float* s0     = (float*)d_ws;                       // NSELF * H1
    float* spikes = s0     + (size_t)NSELF * H1;        // N0 * KHEAD
    float* WrT0   = spikes + (size_t)N0 * KHEAD;        // H1 * F
    float* WlT0   = WrT0   + (size_t)H1 * F;
    float* WrT1   = WlT0   + (size_t)H1 * F;
    float* WlT1   = WrT1   + (size_t)H2 * H1;
    float* WpT    = WlT1   + (size_t)H2 * H1;           // OUTC * KHEAD

    // transpose the (tiny) weights once per launch
    transpose_kernel<<<(F * H1 + 255) / 256, 256, 0, stream>>>(Wr0, WrT0, F, H1);
    transpose_kernel<<<(F * H1 + 255) / 256, 256, 0, stream>>>(Wl0, WlT0, F, H1);
    transpose_kernel<<<(H1 * H2 + 255) / 256, 256, 0, stream>>>(Wr1, WrT1, H1, H2);
    transpose_kernel<<<(H1 * H2 + 255) / 256, 256, 0, stream>>>(Wl1, WlT1, H1, H2);
    transpose_kernel<<<(KHEAD * OUTC + 255) / 256, 256, 0, stream>>>(Wp, WpT, KHEAD, OUTC);

    for (int t = 0; t < TSTEPS; ++t) {
        layer0_kernel<<<NSELF / 32, 256, 0, stream>>>(
            x, nodes,
            nbr1 + (size_t)t * N1,
            nbr2 + (size_t)t * N1 * S2,
            WrT0, WlT0, br0, bl0, s0);
        layer1_kernel<<<N0 / 32, 256, 0, stream>>>(
            s0, WrT1, WlT1, br1, bl1, spikes, t);
    }
    head_kernel<<<N0 / 16, 128, 0, stream>>>(spikes, WpT, bp, out);
}